// ContextInnerProd_2130303779438
// MI455X (gfx1250) — compile-verified
//
#include <hip/hip_runtime.h>
#include <hip/hip_bf16.h>
#include <cstddef>

// Problem sizes (fixed by the reference).
#define S 16384
#define H 2048

#define KSPLIT 4          // K-dimension split for the scores GEMV
#define KCH (H / KSPLIT)  // 512 columns per partial
#define NSC 32            // S-dimension split for the context pass
#define SCH (S / NSC)     // 512 rows per partial

typedef __attribute__((ext_vector_type(2))) float v2f;
typedef __attribute__((ext_vector_type(8))) float v8f;

// ---------------------------------------------------------------------------
// Kernel 1: scores_part[p][s] = <enc[s, p*KCH : (p+1)*KCH], dec[...]>
// One wave computes 16 rows via V_WMMA_F32_16X16X4_F32 accumulation.
//   A (16x4 f32): lane L holds enc[row(L%16)][k + 2*(L/16) + {0,1}]  -> b64 load
//   B (4x16 f32): lane L holds dec[k + 2*(L/16) + {0,1}] (bcast over N)
//   D: lane 0 holds scores for rows +0..7 in c[0..7], lane 16 rows +8..15.
// ---------------------------------------------------------------------------
__global__ void cip_scores_kernel(const float* __restrict__ enc,
                                  const float* __restrict__ dec,
                                  float* __restrict__ scores_part) {
  const int lane = threadIdx.x & 31;
  const int wave = threadIdx.x >> 5;
  const int row0 = blockIdx.x * (8 * 16) + wave * 16;   // 8 waves/block, 16 rows/wave
  const int r    = lane & 15;
  const int koff = (lane >> 4) << 1;                    // 0 for lanes 0-15, 2 for 16-31

  const float* arow = enc + (size_t)(row0 + r) * H + koff;
  const float* bvec = dec + koff;
  const int kbase = blockIdx.y * KCH;

  v8f c = {};
#pragma unroll 4
  for (int k = kbase; k < kbase + KCH; k += 4) {
    v2f a = *(const v2f*)(arow + k);
    v2f b = *(const v2f*)(bvec + k);
    c = __builtin_amdgcn_wmma_f32_16x16x4_f32(
        /*neg_a=*/false, a, /*neg_b=*/false, b,
        /*c_mod=*/(short)0, c, /*reuse_a=*/false, /*reuse_b=*/false);
  }

  // Column N=0 lives in lanes 0 and 16.
  if ((lane & 15) == 0) {
    const int rbase = row0 + ((lane >> 4) << 3);
    float* dst = scores_part + (size_t)blockIdx.y * S + rbase;
#pragma unroll
    for (int v = 0; v < 8; ++v) dst[v] = c[v];
  }
}

// ---------------------------------------------------------------------------
// Kernel 2: stable softmax over S scores (sum KSPLIT partials deterministically,
// max-reduce, exp, sum-reduce, normalize). Single block; 64 KB of traffic.
// ---------------------------------------------------------------------------
__global__ void cip_softmax_kernel(const float* __restrict__ scores_part,
                                   float* __restrict__ weights) {
  __shared__ float red[256];
  const int t = threadIdx.x;

  float lmax = -3.402823466e38f;
  for (int i = t; i < S; i += 256) {
    float s = scores_part[i] + scores_part[S + i] +
              scores_part[2 * S + i] + scores_part[3 * S + i];
    weights[i] = s;                 // stash combined score
    lmax = fmaxf(lmax, s);
  }
  red[t] = lmax;
  __syncthreads();
  for (int off = 128; off > 0; off >>= 1) {
    if (t < off) red[t] = fmaxf(red[t], red[t + off]);
    __syncthreads();
  }
  const float gmax = red[0];
  __syncthreads();

  float lsum = 0.f;
  for (int i = t; i < S; i += 256) {
    float e = __expf(weights[i] - gmax);
    weights[i] = e;
    lsum += e;
  }
  red[t] = lsum;
  __syncthreads();
  for (int off = 128; off > 0; off >>= 1) {
    if (t < off) red[t] += red[t + off];
    __syncthreads();
  }
  const float inv = 1.f / red[0];
  __syncthreads();

  for (int i = t; i < S; i += 256) weights[i] *= inv;
}

// ---------------------------------------------------------------------------
// Kernel 3: ctx_part[chunk][h] = sum over 512 rows of w[s]*enc[s][h].
// Column-parallel, fully coalesced (256 lanes read 1KB contiguous per row).
// ---------------------------------------------------------------------------
__global__ void cip_context_kernel(const float* __restrict__ enc,
                                   const float* __restrict__ weights,
                                   float* __restrict__ ctx_part) {
  const int h  = blockIdx.x * 256 + threadIdx.x;
  const int s0 = blockIdx.y * SCH;
  const float* p = enc + (size_t)s0 * H + h;

  float acc = 0.f;
#pragma unroll 4
  for (int s = 0; s < SCH; ++s) {
    acc = fmaf(weights[s0 + s], p[(size_t)s * H], acc);
  }
  ctx_part[(size_t)blockIdx.y * H + h] = acc;
}

// ---------------------------------------------------------------------------
// Kernel 4: deterministic fixed-order reduction of the NSC context partials.
// ---------------------------------------------------------------------------
__global__ void cip_reduce_kernel(const float* __restrict__ ctx_part,
                                  float* __restrict__ out) {
  const int h = blockIdx.x * 256 + threadIdx.x;
  float acc = 0.f;
#pragma unroll
  for (int p = 0; p < NSC; ++p) acc += ctx_part[(size_t)p * H + h];
  out[h] = acc;
}

// ---------------------------------------------------------------------------
extern "C" void kernel_launch(void* const* d_in, const int* in_sizes, int n_in,
                              void* d_out, int out_size, void* d_ws, size_t ws_size,
                              hipStream_t stream) {
  (void)in_sizes; (void)n_in; (void)out_size; (void)ws_size;

  const float* enc = (const float*)d_in[0];  // [S, H] fp32
  const float* dec = (const float*)d_in[1];  // [1, H] fp32
  float* out = (float*)d_out;                // [H] fp32

  float* ws          = (float*)d_ws;
  float* scores_part = ws;                   // KSPLIT * S floats (256 KB)
  float* weights     = ws + (size_t)KSPLIT * S;        // S floats (64 KB)
  float* ctx_part    = weights + S;                    // NSC * H floats (256 KB)

  // Pass 1: scores GEMV via WMMA. 128 x 4 blocks, 8 waves each.
  cip_scores_kernel<<<dim3(S / 128, KSPLIT), 256, 0, stream>>>(enc, dec, scores_part);

  // Pass 2: softmax (single block; trivial cost).
  cip_softmax_kernel<<<1, 256, 0, stream>>>(scores_part, weights);

  // Pass 3: weighted column reduction (reads enc again, largely L2-resident).
  cip_context_kernel<<<dim3(H / 256, NSC), 256, 0, stream>>>(enc, weights, ctx_part);

  // Pass 4: combine partials into the final [1, H] context.
  cip_reduce_kernel<<<dim3(H / 256), 256, 0, stream>>>(ctx_part, out);
}